// GATV2_11424613007589
// MI455X (gfx1250) — compile-verified
//
#include <hip/hip_runtime.h>
#include <hip/hip_bf16.h>

typedef __attribute__((ext_vector_type(2))) float v2f;
typedef __attribute__((ext_vector_type(8))) float v8f;

#define NEG_SLOPE 0.2f

__device__ __forceinline__ unsigned fenc(float f) {
  unsigned u = __float_as_uint(f);
  return (u & 0x80000000u) ? ~u : (u | 0x80000000u);
}
__device__ __forceinline__ float fdec(unsigned u) {
  return (u & 0x80000000u) ? __uint_as_float(u & 0x7FFFFFFFu) : __uint_as_float(~u);
}

// ---------------------------------------------------------------------------
// Dual GEMM: Yl = X @ Wl, Yr = X @ Wr  (X: [N,K] row-major, W: [K,Ncol] row-major)
// One wave (32 lanes) computes one 16x16 f32 output tile with K/4 WMMA ops.
// blockIdx.z selects Wl->Yl (0) or Wr->Yr (1).
// A frag (16x4 f32, 2 VGPR): lanes 0-15 M=0..15 hold K=k,k+1; lanes 16-31 hold K=k+2,k+3.
// B frag (4x16 f32, 2 VGPR): mirrored K split, N = lane&15.
// D (16x16 f32, 8 VGPR): VGPR r -> (M = r + 8*(lane>>4), N = lane&15).
// ---------------------------------------------------------------------------
__global__ void gemm16_wmma(const float* __restrict__ X,
                            const float* __restrict__ Wl,
                            const float* __restrict__ Wr,
                            float* __restrict__ Yl,
                            float* __restrict__ Yr,
                            int K, int Ncol) {
  const float* W = blockIdx.z ? Wr : Wl;
  float* Y = blockIdx.z ? Yr : Yl;
  const int lane = threadIdx.x & 31;
  const int half = lane >> 4;
  const int lm = lane & 15;
  const int row = blockIdx.x * 16 + lm;
  const int colbase = blockIdx.y * 16;
  const int col = colbase + lm;
  const int colc = col < Ncol ? col : (Ncol - 1);   // clamp loads, mask stores

  v8f acc = {0.f, 0.f, 0.f, 0.f, 0.f, 0.f, 0.f, 0.f};
  const float* xrow = X + (size_t)row * K;

#pragma unroll 8
  for (int k = 0; k < K; k += 4) {
    const int kk = k + half * 2;
    v2f a = *(const v2f*)(xrow + kk);               // K=kk, kk+1 (8B aligned)
    v2f b;
    b.x = W[(size_t)kk * Ncol + colc];              // K=kk   row of B
    b.y = W[(size_t)(kk + 1) * Ncol + colc];        // K=kk+1 row of B
    acc = __builtin_amdgcn_wmma_f32_16x16x4_f32(
        /*neg_a=*/false, a, /*neg_b=*/false, b,
        /*c_mod=*/(short)0, acc, /*reuse_a=*/false, /*reuse_b=*/false);
  }

#pragma unroll
  for (int r = 0; r < 8; ++r) {
    const int rr = blockIdx.x * 16 + r + half * 8;
    const int cc = colbase + lm;
    if (cc < Ncol) Y[(size_t)rr * Ncol + cc] = acc[r];
  }
}

// ---------------------------------------------------------------------------
// Zero/neg-inf initialization of per-node softmax state + output accumulator.
// ---------------------------------------------------------------------------
__global__ void init_state(unsigned* __restrict__ lmax, float* __restrict__ den,
                           float* __restrict__ acc, int nH, int nHC) {
  const int i = blockIdx.x * blockDim.x + threadIdx.x;
  if (i < nH) {
    lmax[i] = 0x007FFFFFu;  // fenc(-inf)
    den[i] = 0.f;
  }
  if (i < nHC) acc[i] = 0.f;
}

// ---------------------------------------------------------------------------
// Pass A: per (edge, head) attention logit + segment max via ordered-uint atomicMax.
// logits[e,h] = sum_c leaky_relu(xl[src,h,c] + xr[dst,h,c]) * att[h,c]
// ---------------------------------------------------------------------------
__global__ void edge_logits(const float* __restrict__ xl, const float* __restrict__ xr,
                            const int* __restrict__ src, const int* __restrict__ dst,
                            const float* __restrict__ att,
                            float* __restrict__ logits, unsigned* __restrict__ lmax,
                            int E, int H, int C) {
  const int i = blockIdx.x * blockDim.x + threadIdx.x;
  if (i >= E * H) return;
  const int e = i / H;
  const int h = i - e * H;
  const int s = src[e];
  const int d = dst[e];
  const float* pl = xl + ((size_t)s * H + h) * C;
  const float* pr = xr + ((size_t)d * H + h) * C;
  const float* pa = att + h * C;
  float v = 0.f;
  for (int c = 0; c < C; ++c) {
    float t = pl[c] + pr[c];
    t = t > 0.f ? t : NEG_SLOPE * t;
    v += t * pa[c];
  }
  logits[i] = v;
  atomicMax(&lmax[(size_t)d * H + h], fenc(v));
}

// ---------------------------------------------------------------------------
// Pass B: p = exp(logit - segmax), accumulate softmax denominator per (node, head).
// p overwrites the logits buffer.
// ---------------------------------------------------------------------------
__global__ void edge_softnum(const int* __restrict__ dst, float* __restrict__ logits,
                             const unsigned* __restrict__ lmax, float* __restrict__ den,
                             int E, int H) {
  const int i = blockIdx.x * blockDim.x + threadIdx.x;
  if (i >= E * H) return;
  const int e = i / H;
  const int h = i - e * H;
  const int d = dst[e];
  const float m = fdec(lmax[(size_t)d * H + h]);
  const float p = expf(logits[i] - m);
  logits[i] = p;
  atomicAdd(&den[(size_t)d * H + h], p);
}

// ---------------------------------------------------------------------------
// Pass C: unnormalized weighted scatter: acc[dst, j] += p[e, j/C] * xl[src, j].
// One thread per (edge, channel): contiguous reads, coalesced atomics.
// Normalization by den happens once per node in node_final (same math as alpha).
// ---------------------------------------------------------------------------
__global__ void edge_scatter(const float* __restrict__ xl, const int* __restrict__ src,
                             const int* __restrict__ dst, const float* __restrict__ p,
                             float* __restrict__ acc, int E, int H, int C) {
  const long long t = (long long)blockIdx.x * blockDim.x + threadIdx.x;
  const int HC = H * C;
  if (t >= (long long)E * HC) return;
  const int e = (int)(t / HC);
  const int j = (int)(t - (long long)e * HC);
  const int h = j / C;
  const int s = src[e];
  const int d = dst[e];
  const float w = p[(size_t)e * H + h];
  atomicAdd(&acc[(size_t)d * HC + j], w * xl[(size_t)s * HC + j]);
}

// ---------------------------------------------------------------------------
// Per-node finalize: divide by softmax denominator, add bias, optional ReLU.
// ---------------------------------------------------------------------------
__global__ void node_final(const float* __restrict__ acc, const float* __restrict__ den,
                           const float* __restrict__ bias, float* __restrict__ out,
                           int Nn, int H, int C, int doRelu) {
  const int i = blockIdx.x * blockDim.x + threadIdx.x;
  const int HC = H * C;
  if (i >= Nn * HC) return;
  const int n = i / HC;
  const int j = i - n * HC;
  const int h = j / C;
  float v = acc[i] / fmaxf(den[(size_t)n * H + h], 1e-16f) + bias[j];
  if (doRelu) v = fmaxf(v, 0.f);
  out[i] = v;
}

// ---------------------------------------------------------------------------
// Host orchestration: 3 GATv2 layers, all on `stream`, scratch in d_ws.
// ---------------------------------------------------------------------------
extern "C" void kernel_launch(void* const* d_in, const int* in_sizes, int n_in,
                              void* d_out, int out_size, void* d_ws, size_t ws_size,
                              hipStream_t stream) {
  const float* x    = (const float*)d_in[0];
  const int* esrc   = (const int*)d_in[1];
  const int* edst   = (const int*)d_in[2];
  const float* Wl0  = (const float*)d_in[3];
  const float* Wr0  = (const float*)d_in[4];
  const float* att0 = (const float*)d_in[5];
  const float* b0   = (const float*)d_in[6];
  const float* Wl1  = (const float*)d_in[7];
  const float* Wr1  = (const float*)d_in[8];
  const float* att1 = (const float*)d_in[9];
  const float* b1   = (const float*)d_in[10];
  const float* Wl2  = (const float*)d_in[11];
  const float* Wr2  = (const float*)d_in[12];
  const float* att2 = (const float*)d_in[13];
  const float* b2   = (const float*)d_in[14];

  const int IN_CH = 128, HID = 128, HEADS = 8, HEAD_DIM = 16;
  const int N = in_sizes[0] / IN_CH;       // 50000 (multiple of 16)
  const int E = in_sizes[1];               // 850000
  const int NUM_CLASSES = in_sizes[14];    // 40

  char* ws = (char*)d_ws;
  size_t off = 0;
  auto wsalloc = [&](size_t bytes) -> void* {
    void* ptr = ws + off;
    off += (bytes + 255) & ~(size_t)255;
    return ptr;
  };
  float* hbuf    = (float*)wsalloc((size_t)N * HID * sizeof(float));
  float* xl      = (float*)wsalloc((size_t)N * HID * sizeof(float));
  float* xr      = (float*)wsalloc((size_t)N * HID * sizeof(float));
  float* pbuf    = (float*)wsalloc((size_t)E * HEADS * sizeof(float));
  unsigned* lmax = (unsigned*)wsalloc((size_t)N * HEADS * sizeof(unsigned));
  float* den     = (float*)wsalloc((size_t)N * HEADS * sizeof(float));
  float* acc     = (float*)wsalloc((size_t)N * HID * sizeof(float));

  auto layer = [&](const float* in, int K, const float* Wl, const float* Wr,
                   const float* att, const float* bias, int H, int C,
                   float* outp, int doRelu) {
    const int HC = H * C;
    const int colTiles = (HC + 15) / 16;

    dim3 gg(N / 16, colTiles, 2);
    gemm16_wmma<<<gg, 32, 0, stream>>>(in, Wl, Wr, xl, xr, K, HC);

    const int nH = N * H, nHC = N * HC;
    const int mx = (nHC > nH) ? nHC : nH;
    init_state<<<(mx + 255) / 256, 256, 0, stream>>>(lmax, den, acc, nH, nHC);

    const int EH = E * H;
    edge_logits<<<(EH + 255) / 256, 256, 0, stream>>>(xl, xr, esrc, edst, att,
                                                      pbuf, lmax, E, H, C);
    edge_softnum<<<(EH + 255) / 256, 256, 0, stream>>>(edst, pbuf, lmax, den, E, H);

    const long long tot = (long long)E * HC;
    const int blks = (int)((tot + 255) / 256);
    edge_scatter<<<blks, 256, 0, stream>>>(xl, esrc, edst, pbuf, acc, E, H, C);

    node_final<<<(nHC + 255) / 256, 256, 0, stream>>>(acc, den, bias, outp,
                                                      N, H, C, doRelu);
  };

  // layer 0: 128 -> 8x16, concat, ReLU
  layer(x, IN_CH, Wl0, Wr0, att0, b0, HEADS, HEAD_DIM, hbuf, 1);
  // layer 1: 128 -> 8x16, concat, ReLU (in-place on hbuf is safe: input only
  // consumed by the GEMM at the start of the layer)
  layer(hbuf, HID, Wl1, Wr1, att1, b1, HEADS, HEAD_DIM, hbuf, 1);
  // layer 2: 128 -> 1x40, mean over 1 head == identity, no ReLU
  layer(hbuf, HID, Wl2, Wr2, att2, b2, 1, NUM_CLASSES, (float*)d_out, 0);
}